// MambaLinearProbe_62345745269515
// MI455X (gfx1250) — compile-verified
//
#include <hip/hip_runtime.h>
#include <hip/hip_bf16.h>
#include <cstdint>

// ---- model dims (match reference) ----
#define V_    32000
#define D_    1024
#define NL_   2
#define DI_   2048
#define N_    16
#define DTR_  64
#define KC_   4
#define B_    4
#define L_    1024
#define BL_   (B_ * L_)      // 4096 rows
#define OUT_  32000
#define DBLW_ (DTR_ + 2 * N_)   // 96

typedef float v2f __attribute__((ext_vector_type(2)));
typedef float v8f __attribute__((ext_vector_type(8)));

__device__ __forceinline__ float sigmoidf_(float x) { return 1.f / (1.f + expf(-x)); }

// ---- CDNA5 async copy: 16B global -> LDS per lane, tracked by ASYNCcnt ----
__device__ __forceinline__ void async_copy16(void* lds, const void* gptr)
{
    // LDS aperture flat address: low 32 bits are the LDS byte offset.
    const uint32_t lds_off = (uint32_t)(uintptr_t)lds;
    asm volatile("global_load_async_to_lds_b128 %0, %1, off"
                 :: "v"(lds_off), "v"(gptr)
                 : "memory");
}
__device__ __forceinline__ void wait_async0()
{
    asm volatile("s_wait_asynccnt 0x0" ::: "memory");
}

// ======================================================================
// Generic fp32 WMMA GEMM:  C[M,N] = A[M,K] @ B[K,N]  (+ residual)
// Block tile 128x64, 8 waves; each wave owns a 16-row strip, 4 n-tiles.
// V_WMMA_F32_16X16X4_F32 compute; double-buffered async-to-LDS staging.
// Requires: M % 128 == 0, K % 16 == 0, N % 16 == 0 (partial 64-col blocks OK).
// ======================================================================
#define BM_ 128
#define BN_ 64
#define KB_ 16

template <bool TRANSB, bool RESID>
__global__ __launch_bounds__(256) void gemm_wmma_f32(
    const float* __restrict__ A, int lda,
    const float* __restrict__ Bm, int ldb,
    float* __restrict__ C, int ldc,
    const float* __restrict__ Rs,
    int M, int N, int K)
{
    // A panels: [buf][128][20]  (stride 20 floats: 16B-aligned rows, conflict-free)
    __shared__ float As[2][BM_][20];
    // B panels: !TRANSB -> k-major [16][64]; TRANSB -> n-major [64][20]
    __shared__ float Braw[2][1280];

    const int tid   = threadIdx.x;
    const int wave  = tid >> 5;
    const int lane  = tid & 31;
    const int mrow  = lane & 15;
    const int khalf = lane >> 4;
    const int row0  = blockIdx.y * BM_;
    const int col0  = blockIdx.x * BN_;

    v8f acc[4];
#pragma unroll
    for (int j = 0; j < 4; ++j)
#pragma unroll
        for (int r = 0; r < 8; ++r) acc[j][r] = 0.f;

    // per-thread staging coordinates
    const int ar  = tid >> 2;          // 0..63 (A row; +64 for second half)
    const int ak4 = (tid & 3) << 2;    // 0,4,8,12

    auto stage = [&](int buf, int k0) {
        // A panel: 128 rows x 16 k, 16B per lane per issue
        async_copy16(&As[buf][ar][ak4],
                     A + (size_t)(row0 + ar) * lda + k0 + ak4);
        async_copy16(&As[buf][ar + 64][ak4],
                     A + (size_t)(row0 + ar + 64) * lda + k0 + ak4);
        if (!TRANSB) {
            // 16 k x 64 n, contiguous along n
            const int k  = tid >> 4;          // 0..15
            const int n4 = (tid & 15) << 2;   // 0..60
            if (col0 + n4 < N)                // N % 16 == 0 => whole f4 in-range
                async_copy16(&Braw[buf][k * BN_ + n4],
                             Bm + (size_t)(k0 + k) * ldb + col0 + n4);
        } else {
            // 64 n x 16 k, contiguous along k (weight rows), n-major LDS
            const int n  = tid >> 2;          // 0..63
            const int k4 = (tid & 3) << 2;    // 0,4,8,12
            if (col0 + n < N)
                async_copy16(&Braw[buf][n * 20 + k4],
                             Bm + (size_t)(col0 + n) * ldb + k0 + k4);
        }
    };

    stage(0, 0);

    int buf = 0;
    for (int k0 = 0; k0 < K; k0 += KB_, buf ^= 1) {
        wait_async0();      // this wave's panel loads have landed in LDS
        __syncthreads();    // all waves' panels visible; previous buffer drained
        if (k0 + KB_ < K) stage(buf ^ 1, k0 + KB_);   // prefetch next panel

#pragma unroll
        for (int kk = 0; kk < KB_; kk += 4) {
            // A frag 16x4: lanes 0-15 hold K=kk,kk+1 ; lanes 16-31 hold K=kk+2,kk+3
            const v2f a = *(const v2f*)&As[buf][wave * 16 + mrow][kk + khalf * 2];
#pragma unroll
            for (int j = 0; j < 4; ++j) {
                v2f b;
                if (TRANSB) {
                    b = *(const v2f*)&Braw[buf][(j * 16 + mrow) * 20 + kk + khalf * 2];
                } else {
                    b.x = Braw[buf][(kk + khalf * 2 + 0) * BN_ + j * 16 + mrow];
                    b.y = Braw[buf][(kk + khalf * 2 + 1) * BN_ + j * 16 + mrow];
                }
                acc[j] = __builtin_amdgcn_wmma_f32_16x16x4_f32(
                    false, a, false, b, (short)0, acc[j], false, false);
            }
        }
        __syncthreads();    // readers done before next iteration's prefetch reuses buf
    }

    // ---- store: VGPR r -> M = r + 8*khalf, N = mrow ----
    const int wrow = row0 + wave * 16;
#pragma unroll
    for (int j = 0; j < 4; ++j) {
        const int nb = col0 + j * 16;
        if (nb + 16 > N) continue;           // uniform per wave
#pragma unroll
        for (int r = 0; r < 8; ++r) {
            const int row = wrow + r + 8 * khalf;
            const int cn  = nb + mrow;
            float val = acc[j][r];
            if (RESID) val += Rs[(size_t)row * ldc + cn];
            C[(size_t)row * ldc + cn] = val;
        }
    }
}

// ======================================================================
// Elementwise / support kernels
// ======================================================================
__global__ __launch_bounds__(256) void embed_kernel(
    const int* __restrict__ ids, const float* __restrict__ emb, float* __restrict__ h)
{
    const int row = blockIdx.x;          // 0..BL-1
    const int t   = threadIdx.x;         // 256 threads x float4 = 1024 floats
    const int id  = ids[row];
    const float4* s = (const float4*)(emb + (size_t)id * D_);
    float4*       d = (float4*)(h + (size_t)row * D_);
    d[t] = s[t];
}

__global__ __launch_bounds__(256) void rmsnorm_kernel(
    const float* __restrict__ x, const float* __restrict__ w, float* __restrict__ out)
{
    __shared__ float red[256];
    const int row = blockIdx.x;
    const int t   = threadIdx.x;
    const float4 v = ((const float4*)(x + (size_t)row * D_))[t];
    red[t] = v.x * v.x + v.y * v.y + v.z * v.z + v.w * v.w;
    __syncthreads();
    for (int s = 128; s > 0; s >>= 1) {
        if (t < s) red[t] += red[t + s];
        __syncthreads();
    }
    const float inv = rsqrtf(red[0] / (float)D_ + 1e-5f);
    const float4 wv = ((const float4*)w)[t];
    float4 o;
    o.x = v.x * inv * wv.x; o.y = v.y * inv * wv.y;
    o.z = v.z * inv * wv.z; o.w = v.w * inv * wv.w;
    ((float4*)(out + (size_t)row * D_))[t] = o;
}

// xc[b,t,c] = silu(conv_b[c] + sum_j xi[b, t+j-3, c] * conv_w[c,j]);  xi = xz[..., :DI]
__global__ __launch_bounds__(256) void conv_silu_kernel(
    const float* __restrict__ xz, const float* __restrict__ cw,
    const float* __restrict__ cb, float* __restrict__ xc)
{
    const int idx = blockIdx.x * 256 + threadIdx.x;    // over BL*DI
    const int c = idx % DI_;
    const int t = (idx / DI_) % L_;
    const int b = idx / (DI_ * L_);
    float acc = cb[c];
#pragma unroll
    for (int j = 0; j < KC_; ++j) {
        const int tt = t + j - (KC_ - 1);
        if (tt >= 0)
            acc += xz[((size_t)(b * L_ + tt)) * (2 * DI_) + c] * cw[c * KC_ + j];
    }
    xc[idx] = acc * sigmoidf_(acc);
}

// delta = softplus(delta + dt_b[c])  (in place)
__global__ __launch_bounds__(256) void softplus_kernel(
    float* __restrict__ delta, const float* __restrict__ dtb)
{
    const int idx = blockIdx.x * 256 + threadIdx.x;
    const int c = idx % DI_;
    const float x = delta[idx] + dtb[c];
    delta[idx] = (x > 20.f) ? x : log1pf(expf(x));
}

// Selective scan: per (b,d) channel, 16-state recurrence in registers; Bt/Ct via LDS.
__global__ __launch_bounds__(512) void scan_kernel(
    const float* __restrict__ delta, const float* __restrict__ xc,
    const float* __restrict__ dbl, const float* __restrict__ A_log,
    float* __restrict__ ys)
{
    const int b = blockIdx.y;
    const int d = blockIdx.x * 512 + threadIdx.x;
    __shared__ float BC[32];                 // Bt[0:16], Ct[16:32]
    float Aa[N_], s[N_];
#pragma unroll
    for (int n = 0; n < N_; ++n) {
        Aa[n] = -expf(A_log[(size_t)d * N_ + n]);
        s[n] = 0.f;
    }
    for (int t = 0; t < L_; ++t) {
        const size_t rowb = (size_t)(b * L_ + t);
        if (threadIdx.x < 32)
            BC[threadIdx.x] = dbl[rowb * DBLW_ + DTR_ + threadIdx.x];
        __syncthreads();
        const size_t idx = rowb * DI_ + d;
        const float dt  = delta[idx];
        const float xv  = xc[idx];
        const float dtx = dt * xv;
        float y = 0.f;
#pragma unroll
        for (int n = 0; n < N_; ++n) {
            s[n] = expf(dt * Aa[n]) * s[n] + dtx * BC[n];
            y += s[n] * BC[16 + n];
        }
        ys[idx] = y;
        __syncthreads();
    }
}

// g = (ys + D_skip[c]*xc) * silu(z),  z = xz[..., DI:2DI]
__global__ __launch_bounds__(256) void gate_kernel(
    const float* __restrict__ ys, const float* __restrict__ xc,
    const float* __restrict__ xz, const float* __restrict__ dskip,
    float* __restrict__ g)
{
    const int idx = blockIdx.x * 256 + threadIdx.x;
    const int c = idx % DI_;
    const size_t row = (size_t)(idx / DI_);
    const float z = xz[row * (2 * DI_) + DI_ + c];
    g[idx] = (ys[idx] + dskip[c] * xc[idx]) * (z * sigmoidf_(z));
}

// ======================================================================
// Host launch
// ======================================================================
extern "C" void kernel_launch(void* const* d_in, const int* in_sizes, int n_in,
                              void* d_out, int out_size, void* d_ws, size_t ws_size,
                              hipStream_t stream)
{
    const int*   ids   = (const int*)d_in[0];
    const float* emb   = (const float*)d_in[1];
    const float* normw = (const float*)d_in[2];
    const float* inpj  = (const float*)d_in[3];
    const float* convw = (const float*)d_in[4];
    const float* convb = (const float*)d_in[5];
    const float* xproj = (const float*)d_in[6];
    const float* dtw   = (const float*)d_in[7];
    const float* dtb   = (const float*)d_in[8];
    const float* alog  = (const float*)d_in[9];
    const float* dskip = (const float*)d_in[10];
    const float* outpj = (const float*)d_in[11];
    const float* fnw   = (const float*)d_in[12];
    const float* headw = (const float*)d_in[13];
    float* out = (float*)d_out;

    float* ws    = (float*)d_ws;
    float* h     = ws;                                   // [BL, D]
    float* xn    = h     + (size_t)BL_ * D_;             // [BL, D]
    float* xz    = xn    + (size_t)BL_ * D_;             // [BL, 2*DI]
    float* xc    = xz    + (size_t)BL_ * 2 * DI_;        // [BL, DI]
    float* dbl   = xc    + (size_t)BL_ * DI_;            // [BL, 96]
    float* delta = dbl   + (size_t)BL_ * DBLW_;          // [BL, DI] (also reused as g)
    float* ysb   = delta + (size_t)BL_ * DI_;            // [BL, DI]

    embed_kernel<<<BL_, 256, 0, stream>>>(ids, emb, h);

    for (int l = 0; l < NL_; ++l) {
        const float* nw = normw + (size_t)l * D_;
        const float* ip = inpj  + (size_t)l * D_ * 2 * DI_;
        const float* cw = convw + (size_t)l * DI_ * KC_;
        const float* cb = convb + (size_t)l * DI_;
        const float* xp = xproj + (size_t)l * DI_ * DBLW_;
        const float* dw = dtw   + (size_t)l * DTR_ * DI_;
        const float* db = dtb   + (size_t)l * DI_;
        const float* al = alog  + (size_t)l * DI_ * N_;
        const float* ds = dskip + (size_t)l * DI_;
        const float* op = outpj + (size_t)l * DI_ * D_;

        // xn = rmsnorm(h)
        rmsnorm_kernel<<<BL_, 256, 0, stream>>>(h, nw, xn);
        // xz = xn @ in_proj          [4096,1024]x[1024,4096]
        gemm_wmma_f32<false, false><<<dim3(2 * DI_ / BN_, BL_ / BM_), 256, 0, stream>>>(
            xn, D_, ip, 2 * DI_, xz, 2 * DI_, nullptr, BL_, 2 * DI_, D_);
        // xc = silu(conv1d(xi))
        conv_silu_kernel<<<BL_ * DI_ / 256, 256, 0, stream>>>(xz, cw, cb, xc);
        // dbl = xc @ x_proj          [4096,2048]x[2048,96]
        gemm_wmma_f32<false, false><<<dim3((DBLW_ + BN_ - 1) / BN_, BL_ / BM_), 256, 0, stream>>>(
            xc, DI_, xp, DBLW_, dbl, DBLW_, nullptr, BL_, DBLW_, DI_);
        // delta_pre = dt_low @ dt_w  [4096,64]x[64,2048]   (dt_low = dbl[:, :64], lda=96)
        gemm_wmma_f32<false, false><<<dim3(DI_ / BN_, BL_ / BM_), 256, 0, stream>>>(
            dbl, DBLW_, dw, DI_, delta, DI_, nullptr, BL_, DI_, DTR_);
        // delta = softplus(delta_pre + dt_b)
        softplus_kernel<<<BL_ * DI_ / 256, 256, 0, stream>>>(delta, db);
        // ys = selective_scan(delta, xc, Bt, Ct, A)
        scan_kernel<<<dim3(DI_ / 512, B_), 512, 0, stream>>>(delta, xc, dbl, al, ysb);
        // g = (ys + D_skip*xc) * silu(z)   (written into delta buffer)
        gate_kernel<<<BL_ * DI_ / 256, 256, 0, stream>>>(ysb, xc, xz, ds, delta);
        // h = h + g @ out_proj       [4096,2048]x[2048,1024], residual in place
        gemm_wmma_f32<false, true><<<dim3(D_ / BN_, BL_ / BM_), 256, 0, stream>>>(
            delta, DI_, op, D_, h, D_, h, BL_, D_, DI_);
    }

    // final norm + vocab head
    rmsnorm_kernel<<<BL_, 256, 0, stream>>>(h, fnw, xn);
    // logits = xn @ head_w^T        [4096,1024]x[1024,32000], B transposed (head_w is [OUT, D])
    gemm_wmma_f32<true, false><<<dim3(OUT_ / BN_, BL_ / BM_), 256, 0, stream>>>(
        xn, D_, headw, D_, out, OUT_, nullptr, BL_, OUT_, D_);
}